// InstrModel_87840671138015
// MI455X (gfx1250) — compile-verified
//
#include <hip/hip_runtime.h>

typedef __attribute__((ext_vector_type(2))) float v2f;
typedef __attribute__((ext_vector_type(8))) float v8f;

#define SEQ   2048
#define ZOOM  256
#define FF    512
#define NL    4
#define KEEP  (SEQ - ZOOM)

// One persistent workgroup (8 wave32s) runs the entire 4-layer model out of LDS.
// The attention branch folds to constants (softmax(ones) = uniform -> inds and
// new_val depend only on fft); the only heavy op is the per-16-row FF=512
// reduction, done with V_WMMA_F32_16X16X4_F32 using an all-ones B matrix so
// D[m][*] = sum_k A[m][k] independent of B's exact lane layout.
__global__ __launch_bounds__(256)
void fused_instr_model_kernel(const float* __restrict__ src,
                              const float* __restrict__ fft,
                              const float* __restrict__ bn2_w,
                              const float* __restrict__ bn2_b,
                              const float* __restrict__ bn2_m,
                              const float* __restrict__ bn2_v,
                              const float* __restrict__ w1,
                              const float* __restrict__ b1,
                              const float* __restrict__ prelu_a,
                              const float* __restrict__ w2,
                              const float* __restrict__ b2,
                              float* __restrict__ out)
{
    __shared__ float s_x[2][SEQ];   // double-buffered activations
    __shared__ float s_vals[ZOOM];
    __shared__ float s_nv[ZOOM];    // new_val (constant across layers)
    __shared__ float s_w1[FF];
    __shared__ float s_b1[FF];
    __shared__ float s_w2[FF];

    const int tid = threadIdx.x;

    // ---- load src into buffer 0 ----
    for (int i = tid; i < SEQ; i += 256) s_x[0][i] = src[i];

    // ---- inds / vals / new_val (input-independent given fft) ----
    {
        // finalp = softmax(ones) = 1/256 exactly; cumsum is exact in fp32.
        float ind = 4095.0f * ((float)(tid + 1) * (1.0f / 256.0f));
        out[SEQ + tid] = ind;                 // second tuple output: inds
        float idx = ind;
        if (idx > 4095.0f) idx -= 1.0f;       // never fires (max == 4095)
        float fl    = floorf(idx);
        float frac0 = idx - fl;
        if (frac0 == 0.0f) idx = (idx > 0.0f) ? (idx - 0.001f) : (idx + 0.001f);
        fl = floorf(idx);
        float frac = idx - fl;
        float v0 = fft[(int)fl];
        float v1 = fft[(int)ceilf(idx)];
        s_vals[tid] = v0 + frac * (v1 - v0);
    }
    __syncthreads();
    s_nv[tid] = (tid == 0) ? s_vals[0] : (s_vals[tid] - s_vals[tid - 1]);
    __syncthreads();

    const int lane  = tid & 31;
    const int wid   = tid >> 5;
    const int m     = lane & 15;   // A-matrix row this lane feeds
    const int khalf = lane >> 4;   // A-matrix K pair: lanes 0-15 -> K0/1, 16-31 -> K2/3

    for (int l = 0; l < NL; ++l) {
        // stage this layer's FF params into LDS
        for (int i = tid; i < FF; i += 256) {
            s_w1[i] = w1[l * FF + i];
            s_b1[i] = b1[l * FF + i];
            s_w2[i] = w2[l * FF + i];
        }
        const float bw = bn2_w[l], bb = bn2_b[l], bm = bn2_m[l], bv = bn2_v[l];
        const float pa = prelu_a[l], bias2 = b2[l];
        const float stdv = sqrtf(bv + 1e-5f);
        const float rstd = 1.0f / stdv;
        // hoist the un-bn divisions: out = y*(std/w) + (m-b)/w  -> one fmaf/elem
        const float oscale = stdv / bw;
        const float oshift = (bm - bb) / bw;
        const int cb = l & 1, nb = cb ^ 1;
        __syncthreads();

        // 128 tiles of 16 rows, round-robined over 8 waves (uniform control flow)
        for (int t = wid; t < SEQ / 16; t += 8) {
            const int row = t * 16 + m;
            // src4: tail rows replaced by constant new_val before bn2
            float x4 = (row < KEEP) ? s_x[cb][row] : s_nv[row - KEEP];
            float s  = (x4 - bm) * rstd * bw + bb;   // bn2

            v8f c = {0.f, 0.f, 0.f, 0.f, 0.f, 0.f, 0.f, 0.f};
            v2f ones = {1.0f, 1.0f};                  // B = all-ones (layout-proof)

            #pragma unroll 4
            for (int f0 = 0; f0 < FF; f0 += 4) {
                const int k0 = f0 + khalf * 2;
                float h0 = fmaf(s_w1[k0],     s, s_b1[k0]);
                float h1 = fmaf(s_w1[k0 + 1], s, s_b1[k0 + 1]);
                h0 = (h0 >= 0.f) ? h0 : pa * h0;      // PReLU
                h1 = (h1 >= 0.f) ? h1 : pa * h1;
                v2f a = { h0 * s_w2[k0], h1 * s_w2[k0 + 1] };
                // D[m][n] += sum_k A[m][k] * 1.0  (accumulates dot over FF)
                c = __builtin_amdgcn_wmma_f32_16x16x4_f32(
                        false, a, false, ones, (short)0, c, false, false);
            }

            // C/D layout: vgpr p holds M=p (lanes 0-15) / M=p+8 (lanes 16-31),
            // all N columns identical -> lane 0 owns rows 0-7, lane 16 rows 8-15.
            if (lane == 0 || lane == 16) {
                const int base = t * 16 + khalf * 8;
                #pragma unroll
                for (int j = 0; j < 8; ++j) {
                    float y = c[j] + bias2;
                    s_x[nb][base + j] = fmaf(y, oscale, oshift);
                }
            }
        }
        __syncthreads();
    }

    // after 4 layers (0->1->0->1->0) result is in buffer 0
    for (int i = tid; i < SEQ; i += 256) out[i] = s_x[0][i];
}

extern "C" void kernel_launch(void* const* d_in, const int* in_sizes, int n_in,
                              void* d_out, int out_size, void* d_ws, size_t ws_size,
                              hipStream_t stream) {
    // setup_inputs() dict order: src, fft, then params in insertion order:
    // in_proj_w, in_proj_b, bn1_w, bn1_b, bn1_m, bn1_v,
    // bn2_w, bn2_b, bn2_m, bn2_v, w1, b1, prelu_a, w2, b2
    const float* src   = (const float*)d_in[0];
    const float* fft   = (const float*)d_in[1];
    const float* bn2w  = (const float*)d_in[8];
    const float* bn2b  = (const float*)d_in[9];
    const float* bn2m  = (const float*)d_in[10];
    const float* bn2v  = (const float*)d_in[11];
    const float* w1    = (const float*)d_in[12];
    const float* b1    = (const float*)d_in[13];
    const float* pa    = (const float*)d_in[14];
    const float* w2    = (const float*)d_in[15];
    const float* b2    = (const float*)d_in[16];

    fused_instr_model_kernel<<<1, 256, 0, stream>>>(
        src, fft, bn2w, bn2b, bn2m, bn2v, w1, b1, pa, w2, b2, (float*)d_out);
}